// BoundaryLoss_37529424233337
// MI455X (gfx1250) — compile-verified
//
#include <hip/hip_runtime.h>
#include <hip/hip_bf16.h>

// ---- geometry constants (B=2, C=4, D=H=W=48) -------------------------------
#define VOL   110592   // 48^3 voxels per (b,c) volume
#define DH    2304     // 48*48 lines per volume per axis
#define NVOL  16       // 2 (pos/neg) * B(2) * C(4) distance volumes
#define NVOX  221184   // B * 48^3
#define NPART 864      // combine blocks (NVOX/256)
#define NPAD  896      // NPART padded to a multiple of 64 for the WMMA loop
#define BIG   1.0e9f   // reference's finite "infinity" for squared distances

typedef float v2f __attribute__((ext_vector_type(2)));
typedef float v8f __attribute__((ext_vector_type(8)));

// ---------------------------------------------------------------------------
// Zero the presence flags and the WMMA padding tail of the partials array.
__global__ void init_small_kernel(int* __restrict__ flags,
                                  float* __restrict__ partials) {
    int t = threadIdx.x;
    if (t < 8) flags[t] = 0;
    if (t >= 8 && t < 8 + (NPAD - NPART)) partials[NPART + (t - 8)] = 0.0f;
}

// One thread per (b, voxel). Writes pos/neg indicator volumes for all 4
// classes and accumulates per-(b,c) "class present" flags.
__global__ void init_masks_kernel(const int* __restrict__ target,
                                  float* __restrict__ dist,
                                  int* __restrict__ flags) {
    __shared__ int bf[8];
    if (threadIdx.x < 8) bf[threadIdx.x] = 0;
    __syncthreads();

    int t = blockIdx.x * blockDim.x + threadIdx.x;   // [0, 2*VOL)
    int b = t / VOL;
    int x = t - b * VOL;
    int tgt = target[t];
#pragma unroll
    for (int c = 0; c < 4; ++c) {
        bool m = (tgt == c);
        int vol = b * 4 + c;
        dist[vol * VOL + x]       = m ? 0.0f : BIG;   // pos mask seed
        dist[(vol + 8) * VOL + x] = m ? BIG : 0.0f;   // neg (~pos) mask seed
    }
    if (tgt >= 0 && tgt < 4) atomicOr(&bf[b * 4 + tgt], 1);
    __syncthreads();
    if (threadIdx.x < 8 && bf[threadIdx.x]) atomicOr(&flags[threadIdx.x], 1);
}

// Exact 1D min-plus squared-distance pass along one axis, in place.
// 4 lines per 256-thread block; 64-thread group per line, 48 lanes active.
__global__ void minplus_kernel(float* __restrict__ dist, int stride) {
    int group = threadIdx.x >> 6;
    int lane  = threadIdx.x & 63;
    int line  = blockIdx.x * 4 + group;              // [0, NVOL*DH)

    int vol = line / DH;
    int r   = line - vol * DH;
    int base;
    if (stride == 1)       base = r * 48;                      // W lines
    else if (stride == 48) base = (r / 48) * DH + (r % 48);    // H lines
    else                   base = r;                           // D lines
    base += vol * VOL;

    __shared__ float lf[4][48];
    if (lane < 48) lf[group][lane] = dist[base + lane * stride];
    __syncthreads();
    if (lane < 48) {
        float best = 3.0e9f;
#pragma unroll
        for (int j = 0; j < 48; ++j) {
            float d = (float)(lane - j);
            best = fminf(best, fmaf(d, d, lf[group][j]));
        }
        dist[base + lane * stride] = best;
    }
}

// Per-voxel softmax over C=4, |p_c * sdf_c| accumulation, block reduction.
__global__ void combine_kernel(const float* __restrict__ pred,
                               const float* __restrict__ dist,
                               const int*   __restrict__ flags,
                               float*       __restrict__ partials) {
    int t = blockIdx.x * blockDim.x + threadIdx.x;   // [0, 2*VOL)
    int b = t / VOL;
    int x = t - b * VOL;

    const float* pb = pred + (size_t)b * 4 * VOL + x;
    float p0 = pb[0], p1 = pb[VOL], p2 = pb[2 * VOL], p3 = pb[3 * VOL];
    float m  = fmaxf(fmaxf(p0, p1), fmaxf(p2, p3));
    float e[4];
    e[0] = expf(p0 - m); e[1] = expf(p1 - m);
    e[2] = expf(p2 - m); e[3] = expf(p3 - m);
    float inv = 1.0f / (e[0] + e[1] + e[2] + e[3]);

    float acc = 0.0f;
#pragma unroll
    for (int c = 0; c < 4; ++c) {
        int vol   = b * 4 + c;
        float pos = dist[vol * VOL + x];
        float neg = dist[(vol + 8) * VOL + x];
        float sdf = flags[vol] ? (sqrtf(pos) - sqrtf(neg)) : 0.0f;
        acc += fabsf(e[c] * inv * sdf);
    }

    __shared__ float red[256];
    red[threadIdx.x] = acc;
    __syncthreads();
    for (int s = 128; s > 0; s >>= 1) {
        if (threadIdx.x < s) red[threadIdx.x] += red[threadIdx.x + s];
        __syncthreads();
    }
    if (threadIdx.x == 0) partials[blockIdx.x] = red[0];
}

// Single wave32; EXEC is all-ones (WMMA requirement). Sum NPAD partials on the
// matrix pipe: with B = all-ones, D = A*1 + C accumulates 64 addends per
// v_wmma_f32_16x16x4_f32 into 16 row-sums (replicated over columns); four
// finalize WMMAs fold the row-sums so every D2 element holds the grand total.
// NPAD is a multiple of 64 and the tail is pre-zeroed, so each lane issues one
// unconditional global_load_b64 per WMMA — no EXEC manipulation in the loop.
__global__ void final_wmma_reduce_kernel(const float* __restrict__ partials,
                                         float* __restrict__ out,
                                         float scale) {
    int lane = threadIdx.x;                          // 32 lanes
    const v2f* pv = (const v2f*)partials;            // NPAD/2 pairs, 8B aligned

    v8f c = {0.f, 0.f, 0.f, 0.f, 0.f, 0.f, 0.f, 0.f};
    v2f ones; ones.x = 1.0f; ones.y = 1.0f;

#pragma unroll 2
    for (int it = 0; it < NPAD / 64; ++it) {
        v2f a = pv[it * 32 + lane];                  // global_load_b64
        c = __builtin_amdgcn_wmma_f32_16x16x4_f32(
                false, a, false, ones, (short)0, c, false, false);
    }
    v8f c2 = {0.f, 0.f, 0.f, 0.f, 0.f, 0.f, 0.f, 0.f};
#pragma unroll
    for (int p = 0; p < 4; ++p) {
        v2f a2; a2.x = c[2 * p]; a2.y = c[2 * p + 1];
        c2 = __builtin_amdgcn_wmma_f32_16x16x4_f32(
                false, a2, false, ones, (short)0, c2, false, false);
    }
    if (lane == 0) out[0] = c2[0] * scale;
}

// ---------------------------------------------------------------------------
extern "C" void kernel_launch(void* const* d_in, const int* in_sizes, int n_in,
                              void* d_out, int out_size, void* d_ws, size_t ws_size,
                              hipStream_t stream) {
    const float* pred   = (const float*)d_in[0];   // (2,4,48,48,48) f32
    const int*   target = (const int*)d_in[1];     // (2,48,48,48)
    float*       out    = (float*)d_out;           // scalar

    // workspace layout: 16 distance volumes | 8 flags | NPAD block partials
    char*  wsb      = (char*)d_ws;
    float* dist     = (float*)wsb;                                   // 7,077,888 B
    int*   flags    = (int*)(wsb + (size_t)NVOL * VOL * sizeof(float));
    float* partials = (float*)(wsb + (size_t)NVOL * VOL * sizeof(float) + 64);

    const int vox_blocks  = NVOX / 256;        // 864
    const int line_blocks = (NVOL * DH) / 4;   // 9216

    init_small_kernel<<<1, 64, 0, stream>>>(flags, partials);
    init_masks_kernel<<<vox_blocks, 256, 0, stream>>>(target, dist, flags);

    // exact separable squared-EDT passes (axis order is irrelevant)
    minplus_kernel<<<line_blocks, 256, 0, stream>>>(dist, DH); // along D
    minplus_kernel<<<line_blocks, 256, 0, stream>>>(dist, 48); // along H
    minplus_kernel<<<line_blocks, 256, 0, stream>>>(dist, 1);  // along W

    combine_kernel<<<vox_blocks, 256, 0, stream>>>(pred, dist, flags, partials);
    final_wmma_reduce_kernel<<<1, 32, 0, stream>>>(
        partials, out, 1.0f / 884736.0f);
}